// GCNPolicy_63136019251805
// MI455X (gfx1250) — compile-verified
//
#include <hip/hip_runtime.h>

typedef __attribute__((ext_vector_type(2))) float v2f;
typedef __attribute__((ext_vector_type(8))) float v8f;

#define HID 16

// ---------------------------------------------------------------- degree pass
__global__ void deg_init_kernel(float* __restrict__ deg, int n) {
    int i = blockIdx.x * blockDim.x + threadIdx.x;
    if (i < n) deg[i] = 1.0f;                 // self-loop weight 1
}

__global__ void deg_accum_kernel(const long long* __restrict__ ei,
                                 const float* __restrict__ ew,
                                 float* __restrict__ deg, int E) {
    int e = blockIdx.x * blockDim.x + threadIdx.x;
    if (e < E) {
        int dst = (int)__builtin_nontemporal_load(ei + (size_t)E + e);
        float w = __builtin_nontemporal_load(ew + e);
        atomicAdd(&deg[dst], w);
    }
}

__global__ void dinv_kernel(float* __restrict__ deg, int n) {
    int i = blockIdx.x * blockDim.x + threadIdx.x;
    if (i < n) {
        float d = deg[i];
        deg[i] = d > 0.0f ? rsqrtf(d) : 0.0f;  // in place: deg -> dinv
    }
}

// ------------------------------------------------- WMMA tile GEMM: T = A @ W
// A: [nrows, K] row-major, W: [K, 16] row-major, T: [nrows, 16].
// One wave per 16-row tile; V_WMMA_F32_16X16X4_F32, K stepped by 4.
// Tail handling: clamp the A row (rows are independent in M).
// Epilogue: stage the 16x16 tile in LDS, then 2 coalesced b128 stores/lane.
template <int K>
__global__ __launch_bounds__(128) void gcn_gemm_wmma_kernel(
        const float* __restrict__ A,
        const float* __restrict__ W,
        float* __restrict__ T,
        int nrows) {
    __shared__ float sW[K * HID];
    __shared__ float sOut[4][16 * HID];       // one 1KB tile per wave
    for (int i = threadIdx.x; i < K * HID; i += blockDim.x) sW[i] = W[i];
    __syncthreads();

    const int lane = threadIdx.x & 31;
    const int wave = threadIdx.x >> 5;
    const int tile = blockIdx.x * 4 + wave;
    const int rowBase = tile * 16;
    const int hi  = lane >> 4;                // 0: K=k,k+1   1: K=k+2,k+3
    const int col = lane & 15;

    int am = rowBase + col;                   // A row this lane supplies
    if (am > nrows - 1) am = nrows - 1;       // clamp -> unconditional loads
    // lane reads float pairs (k+2*hi, k+2*hi+1): one b64 load per WMMA
    const v2f* arow2 = reinterpret_cast<const v2f*>(A + (size_t)am * K) + hi;

    v8f c = {};
#pragma unroll 4
    for (int k = 0; k < K; k += 4) {
        v2f a = __builtin_nontemporal_load(arow2 + (k >> 1));  // NT: streamed once
        v2f b;
        b.x = sW[(k + 2 * hi) * HID + col];
        b.y = sW[(k + 2 * hi + 1) * HID + col];
        c = __builtin_amdgcn_wmma_f32_16x16x4_f32(
                /*neg_a=*/false, a, /*neg_b=*/false, b,
                /*c_mod=*/(short)0, c, /*reuse_a=*/false, /*reuse_b=*/false);
    }

    // C/D layout: VGPR r -> row (r + 8*hi), col = lane&15. Stage via LDS.
    float* so = sOut[wave];
#pragma unroll
    for (int r = 0; r < 8; ++r)
        so[(r + 8 * hi) * HID + col] = c[r];
    __syncthreads();

    // tile output is contiguous: T[rowBase*16 ... +255]; 8 floats per lane
    int row = rowBase + (lane >> 1);          // each chunk lies in one row
    if (row < nrows) {
        const float4* s4 = reinterpret_cast<const float4*>(so + lane * 8);
        float4* d4 = reinterpret_cast<float4*>(T + (size_t)rowBase * HID + lane * 8);
        d4[0] = s4[0];
        d4[1] = s4[1];
    }
}

// ----------------------------------------------- scatter: self loop + edges
__global__ void self_init_kernel(const float* __restrict__ dinv,
                                 const float4* __restrict__ T4,
                                 float4* __restrict__ acc4, int n) {
    int i = blockIdx.x * blockDim.x + threadIdx.x;    // n*4 float4 chunks
    if (i < n * 4) {
        float d = dinv[i >> 2];
        float s = d * d;                              // norm_self = dinv*1*dinv
        float4 t = T4[i];
        acc4[i] = make_float4(s * t.x, s * t.y, s * t.z, s * t.w);
    }
}

__global__ void edge_scatter_kernel(const long long* __restrict__ ei,
                                    const float* __restrict__ ew,
                                    const float* __restrict__ dinv,
                                    const float* __restrict__ T,
                                    float* __restrict__ acc, int E) {
    int tid = blockIdx.x * blockDim.x + threadIdx.x;   // E*4 threads
    int e = tid >> 2;
    if (e >= E) return;
    int q = tid & 3;                                   // 4 floats per thread
    int src = (int)__builtin_nontemporal_load(ei + e);
    int dst = (int)__builtin_nontemporal_load(ei + (size_t)E + e);
    float w = __builtin_nontemporal_load(ew + e) * dinv[src] * dinv[dst];
    const float4 v = reinterpret_cast<const float4*>(T + (size_t)src * HID)[q];
    float* ap = acc + (size_t)dst * HID + q * 4;
    atomicAdd(ap + 0, w * v.x);
    atomicAdd(ap + 1, w * v.y);
    atomicAdd(ap + 2, w * v.z);
    atomicAdd(ap + 3, w * v.w);
}

__global__ void bias_relu_kernel(float4* __restrict__ acc4,
                                 const float4* __restrict__ b4, int n) {
    int i = blockIdx.x * blockDim.x + threadIdx.x;    // n*4 float4 chunks
    if (i < n * 4) {
        float4 v = acc4[i];
        float4 b = b4[i & 3];
        v.x = fmaxf(v.x + b.x, 0.0f);
        v.y = fmaxf(v.y + b.y, 0.0f);
        v.z = fmaxf(v.z + b.z, 0.0f);
        v.w = fmaxf(v.w + b.w, 0.0f);
        acc4[i] = v;
    }
}

// -------------------------------------------------------------- pooling+head
__global__ void pool_init_kernel(int* __restrict__ pooled, int total) {
    int i = blockIdx.x * blockDim.x + threadIdx.x;
    if (i < total) pooled[i] = 0;             // 0.0f bits; matches finite-guard
}

__global__ void pool_max_kernel(const float* __restrict__ h,
                                const long long* __restrict__ batch,
                                int* __restrict__ pooled, int n) {
    int i = blockIdx.x * blockDim.x + threadIdx.x;
    if (i < n * HID) {
        int node = i >> 4;
        int g = (int)batch[node];
        // h >= 0 after ReLU: IEEE order == int order for non-negative floats
        atomicMax(&pooled[g * HID + (i & 15)], __float_as_int(h[i]));
    }
}

__global__ void head_kernel(const int* __restrict__ pooled_i,
                            const float* __restrict__ Wl,
                            const float* __restrict__ bl,
                            float* __restrict__ out, int G, int nact) {
    int i = blockIdx.x * blockDim.x + threadIdx.x;
    if (i < G * nact) {
        int g = i / nact, a = i % nact;
        const float* p = reinterpret_cast<const float*>(pooled_i) + g * HID;
        float s = bl[a];
#pragma unroll
        for (int f = 0; f < HID; ++f) s += p[f] * Wl[f * nact + a];
        out[i] = s;
    }
}

// ------------------------------------------------------------------- launch
extern "C" void kernel_launch(void* const* d_in, const int* in_sizes, int n_in,
                              void* d_out, int out_size, void* d_ws, size_t ws_size,
                              hipStream_t stream) {
    const float*     x     = (const float*)d_in[0];
    const long long* ei    = (const long long*)d_in[1];   // int64 [2, E]
    const float*     ew    = (const float*)d_in[2];
    const long long* batch = (const long long*)d_in[3];   // int64 [N]
    const float*     W1    = (const float*)d_in[4];
    const float*     b1    = (const float*)d_in[5];
    const float*     W2    = (const float*)d_in[6];
    const float*     b2    = (const float*)d_in[7];
    const float*     Wl    = (const float*)d_in[8];
    const float*     bl    = (const float*)d_in[9];
    float*           out   = (float*)d_out;

    const int N    = in_sizes[3];             // nodes (batch length)
    const int E    = in_sizes[2];             // edges (edge_weight length)
    const int NACT = in_sizes[9];             // 18
    const int G    = out_size / NACT;         // 64

    // workspace layout
    char* ws = (char*)d_ws;
    size_t off = 0;
    auto alloc = [&](size_t bytes) { void* p = ws + off; off = (off + bytes + 255) & ~(size_t)255; return p; };
    float* dinv   = (float*)alloc((size_t)N * sizeof(float));
    float* tbuf   = (float*)alloc((size_t)N * HID * sizeof(float));
    float* accbuf = (float*)alloc((size_t)N * HID * sizeof(float));
    int*   pooled = (int*)alloc((size_t)G * HID * sizeof(int));
    (void)ws_size; (void)n_in;

    const int TB = 256;
    const int vec4Threads = N * 4;                            // float4 granularity
    const int gemmTiles  = (N + 15) / 16;
    const int gemmBlocks = (gemmTiles + 3) / 4;               // 4 waves/block

    // degrees -> dinv
    deg_init_kernel<<<(N + TB - 1) / TB, TB, 0, stream>>>(dinv, N);
    deg_accum_kernel<<<(E + TB - 1) / TB, TB, 0, stream>>>(ei, ew, dinv, E);
    dinv_kernel<<<(N + TB - 1) / TB, TB, 0, stream>>>(dinv, N);

    // layer 1: t = x @ W1 ; acc = scatter(norm * t) ; h1 = relu(acc + b1)
    gcn_gemm_wmma_kernel<128><<<gemmBlocks, 128, 0, stream>>>(x, W1, tbuf, N);
    self_init_kernel<<<(vec4Threads + TB - 1) / TB, TB, 0, stream>>>(
        dinv, (const float4*)tbuf, (float4*)accbuf, N);
    edge_scatter_kernel<<<((size_t)E * 4 + TB - 1) / TB, TB, 0, stream>>>(ei, ew, dinv, tbuf, accbuf, E);
    bias_relu_kernel<<<(vec4Threads + TB - 1) / TB, TB, 0, stream>>>(
        (float4*)accbuf, (const float4*)b1, N);

    // layer 2: t = h1 @ W2 ; acc = scatter(norm * t) ; h2 = relu(acc + b2)
    gcn_gemm_wmma_kernel<HID><<<gemmBlocks, 128, 0, stream>>>(accbuf, W2, tbuf, N);
    self_init_kernel<<<(vec4Threads + TB - 1) / TB, TB, 0, stream>>>(
        dinv, (const float4*)tbuf, (float4*)accbuf, N);
    edge_scatter_kernel<<<((size_t)E * 4 + TB - 1) / TB, TB, 0, stream>>>(ei, ew, dinv, tbuf, accbuf, E);
    bias_relu_kernel<<<(vec4Threads + TB - 1) / TB, TB, 0, stream>>>(
        (float4*)accbuf, (const float4*)b2, N);

    // global max pool + linear head
    pool_init_kernel<<<(G * HID + TB - 1) / TB, TB, 0, stream>>>(pooled, G * HID);
    pool_max_kernel<<<(N * HID + TB - 1) / TB, TB, 0, stream>>>(accbuf, batch, pooled, N);
    head_kernel<<<(G * NACT + TB - 1) / TB, TB, 0, stream>>>(pooled, Wl, bl, out, G, NACT);
}